// GNN_14388140442154
// MI455X (gfx1250) — compile-verified
//
#include <hip/hip_runtime.h>
#include <hip/hip_bf16.h>

typedef float v2f __attribute__((ext_vector_type(2)));
typedef float v8f __attribute__((ext_vector_type(8)));

// ---------------------------------------------------------------------------
// deg[i] = 1.0 (self loop weight)
__global__ void gcn_init_deg(float* __restrict__ deg, int n) {
    int i = blockIdx.x * blockDim.x + threadIdx.x;
    if (i < n) deg[i] = 1.0f;
}

// deg[dst[e]] += ew[e]
__global__ void gcn_deg_accum(float* __restrict__ deg, const int* __restrict__ ei,
                              const float* __restrict__ ew, int E, int n) {
    int e = blockIdx.x * blockDim.x + threadIdx.x;
    if (e >= E) return;
    int d = ei[E + e];
    if ((unsigned)d < (unsigned)n) atomicAdd(&deg[d], ew[e]);
}

// dis[i] = deg>0 ? rsqrt(deg) : 0   (in place)
__global__ void gcn_dis(float* __restrict__ deg, int n) {
    int i = blockIdx.x * blockDim.x + threadIdx.x;
    if (i >= n) return;
    float v = deg[i];
    deg[i] = (v > 0.0f) ? rsqrtf(fmaxf(v, 1e-12f)) : 0.0f;
}

// ---------------------------------------------------------------------------
// WMMA f32 GEMM with fused self-loop epilogue:
//   h    = f(A) @ W            f = RELU ? relu(a + abias) : a
//   out2 = dis^2 * h (+obias)  (self-loop term of the GCN aggregation)
// Block: 256 threads = 8 waves; wave w owns rows [blk*128 + w*16, +16),
// all OC/16 column tiles.  W staged in LDS pre-swizzled so each lane's B
// fragment {W[2p][col], W[2p+1][col]} is one aligned ds_load_b64.
template <int OC, bool RELU>
__global__ __launch_bounds__(256) void gcn_gemm_wmma(
    const float* __restrict__ A, const float* __restrict__ abias,
    const float* __restrict__ W, const float* __restrict__ dis,
    float* __restrict__ out, float* __restrict__ out2,
    const float* __restrict__ obias, int M) {
    constexpr int K  = 128;
    constexpr int NT = OC / 16;
    extern __shared__ float sW[];  // K * OC floats, pair-swizzled

    // swizzled fill: sW[(k/2)*2*OC + col*2 + (k&1)] = W[k*OC + col]
    const int tid = threadIdx.x;
    for (int i = tid * 4; i < K * OC; i += 256 * 4) {
        const float4 v = *reinterpret_cast<const float4*>(W + i);
        const int k = i / OC, c = i % OC;
        float* dstp = sW + (k >> 1) * 2 * OC + (k & 1);
        dstp[(c + 0) * 2] = v.x;
        dstp[(c + 1) * 2] = v.y;
        dstp[(c + 2) * 2] = v.z;
        dstp[(c + 3) * 2] = v.w;
    }
    __syncthreads();

    const int wave = tid >> 5;
    const int lane = tid & 31;
    const int lh   = lane & 15;
    const int r0   = blockIdx.x * 128 + wave * 16;

    // Clamp A-row so loads stay in bounds without breaking EXEC=all-1s.
    int rowA = r0 + lh;
    if (rowA >= M) rowA = M - 1;
    const float* arow = A + (size_t)rowA * K;
    const int kbase = (lane < 16) ? 0 : 2;  // A/B VGPR0 carries K=0 / K=2
    const int pbase = (lane < 16) ? 0 : 1;  // pair index within k-step

    v8f acc[NT] = {};

    for (int kk = 0; kk < K / 4; ++kk) {
        const int k0 = kk * 4 + kbase;
        v2f a = *reinterpret_cast<const v2f*>(arow + k0);
        if (RELU) {
            v2f bi = *reinterpret_cast<const v2f*>(abias + k0);
            a.x = fmaxf(a.x + bi.x, 0.0f);
            a.y = fmaxf(a.y + bi.y, 0.0f);
        }
        const int pair = kk * 2 + pbase;
#pragma unroll
        for (int ct = 0; ct < NT; ++ct) {
            const int col = ct * 16 + lh;
            v2f b = *reinterpret_cast<const v2f*>(sW + (pair * OC + col) * 2);
            acc[ct] = __builtin_amdgcn_wmma_f32_16x16x4_f32(
                false, a, false, b, (short)0, acc[ct], false, false);
        }
    }

    // C/D layout: vgpr j -> M=j (lanes 0-15) / M=j+8 (lanes 16-31), N=lane%16
#pragma unroll
    for (int j = 0; j < 8; ++j) {
        const int m   = (lane < 16) ? j : j + 8;
        const int row = r0 + m;
        if (row >= M) continue;
        const float dv = dis[row];
        const float d2 = dv * dv;
#pragma unroll
        for (int ct = 0; ct < NT; ++ct) {
            const float  v = acc[ct][j];
            const size_t o = (size_t)row * OC + ct * 16 + lh;
            out[o] = v;
            float s = d2 * v;
            if (obias) s += obias[ct * 16 + lh];
            out2[o] = s;
        }
    }
}

// ---------------------------------------------------------------------------
// agg[dst,c] += dis[src]*ew*dis[dst] * h[src,c]; one thread = (edge, 4 chans)
template <int C>
__global__ void gcn_edge_scatter(float* __restrict__ agg, const float* __restrict__ h,
                                 const float* __restrict__ dis,
                                 const int* __restrict__ ei,
                                 const float* __restrict__ ew, int E, int n) {
    constexpr int CH = C / 4;
    long long t = (long long)blockIdx.x * blockDim.x + threadIdx.x;
    if (t >= (long long)E * CH) return;
    int e  = (int)(t / CH);
    int c4 = (int)(t % CH) * 4;
    int s = ei[e];
    int d = ei[E + e];
    if ((unsigned)s >= (unsigned)n || (unsigned)d >= (unsigned)n) return;
    float w = dis[s] * ew[e] * dis[d];
    float4 hv = *reinterpret_cast<const float4*>(h + (size_t)s * C + c4);
    float* o = agg + (size_t)d * C + c4;
    atomicAdd(o + 0, w * hv.x);
    atomicAdd(o + 1, w * hv.y);
    atomicAdd(o + 2, w * hv.z);
    atomicAdd(o + 3, w * hv.w);
}

// ---------------------------------------------------------------------------
extern "C" void kernel_launch(void* const* d_in, const int* in_sizes, int n_in,
                              void* d_out, int out_size, void* d_ws, size_t ws_size,
                              hipStream_t stream) {
    const float* x  = (const float*)d_in[0];
    const int*   ei = (const int*)d_in[1];   // [2,E] flat: src row then dst row
    const float* ew = (const float*)d_in[2];
    const float* W1 = (const float*)d_in[3];
    const float* b1 = (const float*)d_in[4];
    const float* W2 = (const float*)d_in[5];
    const float* b2 = (const float*)d_in[6];
    float* out = (float*)d_out;

    const int N = in_sizes[0] / 128;
    const int E = in_sizes[2];

    float* ws  = (float*)d_ws;
    float* dis = ws;                       // N (deg, then rsqrt in place)
    float* h   = ws + N;                   // N*128 (h1, later reused as h2 N*64)
    float* agg = h + (size_t)N * 128;      // N*128

    const int TB = 256;
    auto cdiv = [](long long a, long long b) { return (int)((a + b - 1) / b); };

    // normalization coefficients
    gcn_init_deg<<<cdiv(N, TB), TB, 0, stream>>>(dis, N);
    gcn_deg_accum<<<cdiv(E, TB), TB, 0, stream>>>(dis, ei, ew, E, N);
    gcn_dis<<<cdiv(N, TB), TB, 0, stream>>>(dis, N);

    // layer 1: h = x @ W1 (WMMA), agg initialized with dis^2*h in epilogue,
    // then edge scatter completes agg = D^-1/2 A D^-1/2 h
    gcn_gemm_wmma<128, false>
        <<<cdiv(N, 128), 256, 128 * 128 * sizeof(float), stream>>>(
            x, nullptr, W1, dis, h, agg, nullptr, N);
    gcn_edge_scatter<128><<<cdiv((long long)E * 32, TB), TB, 0, stream>>>(
        agg, h, dis, ei, ew, E, N);

    // layer 2: h2 = relu(agg + b1) @ W2 (bias+relu folded into A-load),
    // d_out initialized with dis^2*h2 + b2 in epilogue, then edge scatter
    gcn_gemm_wmma<64, true>
        <<<cdiv(N, 128), 256, 128 * 64 * sizeof(float), stream>>>(
            agg, b1, W2, dis, h, out, b2, N);
    gcn_edge_scatter<64><<<cdiv((long long)E * 16, TB), TB, 0, stream>>>(
        out, h, dis, ei, ew, E, N);
}